// Stage_Wise_Alignment_49203145343362
// MI455X (gfx1250) — compile-verified
//
#include <hip/hip_runtime.h>

typedef float v2f __attribute__((ext_vector_type(2)));
typedef float v8f __attribute__((ext_vector_type(8)));
typedef int   i32x4 __attribute__((vector_size(16)));   // matches builtin param type

#define N_ROWS   4096
#define DIM      256
#define LDS_STRIDE 68   // 68*4 B row pitch: 16B-aligned rows, bank stride 4 -> conflict-free

#if defined(__gfx1250__) && __has_builtin(__builtin_amdgcn_global_load_async_to_lds_b128)
#define ASYNC_BUILTIN 1
#else
#define ASYNC_BUILTIN 0
#endif

// ---- CDNA5 async global->LDS copy (16B per lane), tracked by ASYNCcnt ----
__device__ __forceinline__ void async_cp16(const float* g, float* l) {
#if ASYNC_BUILTIN
    __builtin_amdgcn_global_load_async_to_lds_b128(
        (__attribute__((address_space(1))) i32x4*)(uintptr_t)g,
        (__attribute__((address_space(3))) i32x4*)(__attribute__((address_space(3))) float*)l,
        0, 0);
#else
    unsigned lds_off = (unsigned)(uintptr_t)(__attribute__((address_space(3))) float*)l;
    unsigned long long ga = (unsigned long long)(uintptr_t)g;
    asm volatile("global_load_async_to_lds_b128 %0, %1, off"
                 :: "v"(lds_off), "v"(ga) : "memory");
#endif
}

__device__ __forceinline__ void async_wait0() {
#if defined(__gfx1250__) && __has_builtin(__builtin_amdgcn_s_wait_asynccnt)
    __builtin_amdgcn_s_wait_asynccnt(0);
#else
    asm volatile("s_wait_asynccnt 0" ::: "memory");
#endif
}

__device__ __forceinline__ v8f wmma_f32(const v2f a, const v2f b, v8f c) {
    return __builtin_amdgcn_wmma_f32_16x16x4_f32(false, a, false, b, (short)0, c, false, false);
}

// ---------------- Kernel 1: per-row squared norms + partial column sums ----------------
// grid 64 x 256 threads; block b handles rows [b*128, b*128+128)
__global__ void __launch_bounds__(256) stats_kernel(const float* __restrict__ S,
                                                    const float* __restrict__ T,
                                                    float* __restrict__ sq,
                                                    float* __restrict__ pcol) {
    const int tid = threadIdx.x;   // column 0..255
    const int b   = blockIdx.x;    // 0..63
    __shared__ float wpart[8];
    float cs = 0.f;
    for (int rr = 0; rr < 128; ++rr) {
        const int r = b * 128 + rr;
        const float* row = (r < N_ROWS) ? (S + (size_t)r * DIM)
                                        : (T + (size_t)(r - N_ROWS) * DIM);
        const float x = row[tid];
        cs += x;
        float v = x * x;
        #pragma unroll
        for (int off = 16; off > 0; off >>= 1) v += __shfl_xor(v, off, 32);
        if ((tid & 31) == 0) wpart[tid >> 5] = v;
        __syncthreads();
        if (tid == 0) {
            float s = 0.f;
            #pragma unroll
            for (int w = 0; w < 8; ++w) s += wpart[w];
            sq[r] = s;
        }
        __syncthreads();
    }
    pcol[b * 256 + tid] = cs;
}

// ---------------- Kernel 2: bandwidth from sum(L2) closed form ----------------
__global__ void __launch_bounds__(256) finalize_bw_kernel(const float* __restrict__ sq,
                                                          const float* __restrict__ pcol,
                                                          float* __restrict__ neg_inv5) {
    const int tid = threadIdx.x;
    __shared__ float red[256];
    float cs = 0.f;
    for (int b = 0; b < 64; ++b) cs += pcol[b * 256 + tid];
    red[tid] = cs * cs;
    __syncthreads();
    for (int s = 128; s > 0; s >>= 1) { if (tid < s) red[tid] += red[tid + s]; __syncthreads(); }
    const float ss_tot = red[0];
    __syncthreads();
    float sm = 0.f;
    for (int k = 0; k < 32; ++k) sm += sq[tid + 256 * k];
    red[tid] = sm;
    __syncthreads();
    for (int s = 128; s > 0; s >>= 1) { if (tid < s) red[tid] += red[tid + s]; __syncthreads(); }
    if (tid == 0) {
        const float sumsq = red[0];
        const float n = 2.f * (float)N_ROWS;                    // 8192
        const float sumL2 = 2.f * n * sumsq - 2.f * ss_tot;
        float bw = sumL2 / (n * n - n);
        bw = bw * 0.25f;                                        // / KERNEL_MUL^(KERNEL_NUM//2)
        #pragma unroll
        for (int i = 0; i < 5; ++i) neg_inv5[i] = -1.f / (bw * (float)(1 << i));
    }
}

// ---------------- Kernel 3: fused S*T^T GEMM (WMMA f32 16x16x4) + 5-kernel exp sum ----------------
// grid (32,64), 256 threads = 8 waves. Block tile 64(M) x 128(N), K-blocked by 64 via async LDS.
// Wave w: 32x32 tile at m0w=(w>>2)*32, n0w=(w&3)*32 -> 2x2 accumulators (outer-product reuse).
__global__ void __launch_bounds__(256) mmd_gemm_kernel(const float* __restrict__ S,
                                                       const float* __restrict__ T,
                                                       const float* __restrict__ sq,
                                                       const float* __restrict__ neg_inv5,
                                                       float* __restrict__ partials) {
    __shared__ float lA[64 * LDS_STRIDE];
    __shared__ float lB[128 * LDS_STRIDE];
    __shared__ float red[256];

    const int tid  = threadIdx.x;
    const int wave = tid >> 5;
    const int lane = tid & 31;
    const int r0 = blockIdx.y * 64;     // M block
    const int c0 = blockIdx.x * 128;    // N block
    const int m0w = (wave >> 2) * 32;
    const int n0w = (wave & 3) * 32;

    float ninv[5];
    #pragma unroll
    for (int i = 0; i < 5; ++i) ninv[i] = neg_inv5[i];

    v8f acc00 = {}, acc01 = {}, acc10 = {}, acc11 = {};

    const int lrow  = tid >> 4;         // 0..15
    const int lcol  = (tid & 15) * 4;   // 0..60
    const int khalf = (lane >> 4) * 2;  // which K-pair this half-wave owns
    const int ml    = lane & 15;        // M (for A) / N (for B) within 16-wide tile

    for (int kb = 0; kb < DIM; kb += 64) {
        #pragma unroll
        for (int i = 0; i < 4; ++i) {
            const int rg = lrow + 16 * i;
            async_cp16(S + (size_t)(r0 + rg) * DIM + kb + lcol, &lA[rg * LDS_STRIDE + lcol]);
        }
        #pragma unroll
        for (int i = 0; i < 8; ++i) {
            const int rg = lrow + 16 * i;
            async_cp16(T + (size_t)(c0 + rg) * DIM + kb + lcol, &lB[rg * LDS_STRIDE + lcol]);
        }
        async_wait0();
        __syncthreads();
        #pragma unroll
        for (int k = 0; k < 64; k += 4) {
            const v2f a0 = *(const v2f*)&lA[(m0w      + ml) * LDS_STRIDE + k + khalf];
            const v2f a1 = *(const v2f*)&lA[(m0w + 16 + ml) * LDS_STRIDE + k + khalf];
            const v2f b0 = *(const v2f*)&lB[(n0w      + ml) * LDS_STRIDE + k + khalf];
            const v2f b1 = *(const v2f*)&lB[(n0w + 16 + ml) * LDS_STRIDE + k + khalf];
            acc00 = wmma_f32(a0, b0, acc00);
            acc01 = wmma_f32(a0, b1, acc01);
            acc10 = wmma_f32(a1, b0, acc10);
            acc11 = wmma_f32(a1, b1, acc11);
        }
        __syncthreads();
    }

    // Epilogue: L2 = sq_s[m] + sq_t[n] - 2*gram ; sum_i exp(L2 * -1/(bw*2^i))
    const int hi = lane >> 4;
    const float sqt0 = sq[N_ROWS + c0 + n0w + ml];
    const float sqt1 = sq[N_ROWS + c0 + n0w + 16 + ml];
    float accsum = 0.f;
    #pragma unroll
    for (int v = 0; v < 8; ++v) {
        const int mg0 = r0 + m0w + v + 8 * hi;
        const float sq0 = sq[mg0];
        const float sq1 = sq[mg0 + 16];
        const float L2a = sq0 + sqt0 - 2.f * acc00[v];
        const float L2b = sq0 + sqt1 - 2.f * acc01[v];
        const float L2c = sq1 + sqt0 - 2.f * acc10[v];
        const float L2d = sq1 + sqt1 - 2.f * acc11[v];
        #pragma unroll
        for (int i = 0; i < 5; ++i) {
            accsum += __expf(L2a * ninv[i]);
            accsum += __expf(L2b * ninv[i]);
            accsum += __expf(L2c * ninv[i]);
            accsum += __expf(L2d * ninv[i]);
        }
    }
    red[tid] = accsum;
    __syncthreads();
    for (int s = 128; s > 0; s >>= 1) { if (tid < s) red[tid] += red[tid + s]; __syncthreads(); }
    if (tid == 0) partials[blockIdx.y * gridDim.x + blockIdx.x] = red[0];
}

// ---------------- Kernel 4: deterministic final reduction ----------------
__global__ void __launch_bounds__(256) finish_kernel(const float* __restrict__ partials,
                                                     int nparts,
                                                     float* __restrict__ out) {
    const int tid = threadIdx.x;
    __shared__ float red[256];
    float s = 0.f;
    for (int i = tid; i < nparts; i += 256) s += partials[i];
    red[tid] = s;
    __syncthreads();
    for (int st = 128; st > 0; st >>= 1) { if (tid < st) red[tid] += red[tid + st]; __syncthreads(); }
    if (tid == 0) {
        const float XY = red[0] / ((float)N_ROWS * (float)N_ROWS);
        out[0] = -2.f * XY;   // loss = -XY - YX, and XY == YX by symmetry
    }
}

extern "C" void kernel_launch(void* const* d_in, const int* in_sizes, int n_in,
                              void* d_out, int out_size, void* d_ws, size_t ws_size,
                              hipStream_t stream) {
    const float* S = (const float*)d_in[0];
    const float* T = (const float*)d_in[1];
    float* ws = (float*)d_ws;
    float* sq       = ws;                       // 8192 floats
    float* pcol     = ws + 8192;                // 64*256 = 16384 floats
    float* ninv     = ws + 8192 + 16384;        // 5 floats (+3 pad)
    float* partials = ws + 8192 + 16384 + 8;    // 2048 floats
    float* out = (float*)d_out;

    stats_kernel<<<64, 256, 0, stream>>>(S, T, sq, pcol);
    finalize_bw_kernel<<<1, 256, 0, stream>>>(sq, pcol, ninv);
    mmd_gemm_kernel<<<dim3(32, 64), 256, 0, stream>>>(S, T, sq, ninv, partials);
    finish_kernel<<<1, 256, 0, stream>>>(partials, 32 * 64, out);
}